// CrossSparseGAT_8495445311614
// MI455X (gfx1250) — compile-verified
//
#include <hip/hip_runtime.h>
#include <math.h>

typedef __attribute__((ext_vector_type(2))) float v2f;
typedef __attribute__((ext_vector_type(8))) float v8f;

#define DIM 64
#define NHEAD 8

// ---------------------------------------------------------------------------
// One wave computes a 16-row x 64-col tile of  Out = X @ W.T (+ bias)
// using chained V_WMMA_F32_16X16X4_F32 (fp32 tensor path, K=64 in 16 steps).
// A layout (16x4 f32): lanes 0-15 row m hold K={4k,4k+1}; lanes 16-31 hold
// K={4k+2,4k+3}. B (4x16) mirrors this with W's row-major (out,in) layout.
// D (16x16 f32): VGPR r -> row r (lanes 0-15) / row r+8 (lanes 16-31).
// ---------------------------------------------------------------------------
__device__ __forceinline__ void gemm_tile_16x64(
    const float* __restrict__ X, const float* __restrict__ W,
    float* __restrict__ Out, int r0, int n, const float* __restrict__ bias)
{
    const int lane = threadIdx.x & 31;
    const int half = lane >> 4;
    const int m    = lane & 15;

    int rm = r0 + m; if (rm >= n) rm = n - 1;
    const float* arow = X + (size_t)rm * DIM + half * 2;

    v2f a[16];
#pragma unroll
    for (int kb = 0; kb < 16; ++kb)
        a[kb] = *(const v2f*)(arow + kb * 4);

#pragma unroll
    for (int nb = 0; nb < 4; ++nb) {
        const float* wrow = W + (size_t)(nb * 16 + m) * DIM + half * 2;
        v8f c = {};
#pragma unroll
        for (int kb = 0; kb < 16; ++kb) {
            v2f b = *(const v2f*)(wrow + kb * 4);
            c = __builtin_amdgcn_wmma_f32_16x16x4_f32(
                    false, a[kb], false, b, (short)0, c, false, false);
        }
        float bv = bias ? bias[nb * 16 + m] : 0.0f;
#pragma unroll
        for (int r = 0; r < 8; ++r) {
            int row = r0 + half * 8 + r;
            if (row < n)
                Out[(size_t)row * DIM + nb * 16 + m] = c[r] + bv;
        }
    }
}

// src_proj = src @ W2.T ; V = src @ Wv.T
__global__ void __launch_bounds__(128)
node_proj_src_kernel(const float* __restrict__ src_feats,
                     const float* __restrict__ W2, const float* __restrict__ Wv,
                     float* __restrict__ src_proj, float* __restrict__ V, int n)
{
    int tile = blockIdx.x * (blockDim.x >> 5) + (threadIdx.x >> 5);
    int r0 = tile * 16;
    if (r0 >= n) return;
    gemm_tile_16x64(src_feats, W2, src_proj, r0, n, nullptr);
    gemm_tile_16x64(src_feats, Wv, V, r0, n, nullptr);
}

// dst_proj = dst @ W1.T ; resid = dst @ Wres.T + b_res
__global__ void __launch_bounds__(128)
node_proj_dst_kernel(const float* __restrict__ dst_feats,
                     const float* __restrict__ W1, const float* __restrict__ Wres,
                     const float* __restrict__ b_res,
                     float* __restrict__ dst_proj, float* __restrict__ resid, int n)
{
    int tile = blockIdx.x * (blockDim.x >> 5) + (threadIdx.x >> 5);
    int r0 = tile * 16;
    if (r0 >= n) return;
    gemm_tile_16x64(dst_feats, W1, dst_proj, r0, n, nullptr);
    gemm_tile_16x64(dst_feats, Wres, resid, r0, n, b_res);
}

__global__ void init_kernel(float* __restrict__ agg, int* __restrict__ seg_max_i,
                            float* __restrict__ seg_sum, int n_dst)
{
    int i = blockIdx.x * blockDim.x + threadIdx.x;
    if (i < n_dst * DIM) agg[i] = 0.0f;
    if (i < n_dst * NHEAD) {
        seg_max_i[i] = (int)0x80000000;   // maps below -inf
        seg_sum[i]   = 0.0f;
    }
}

// 8 lanes per edge; lane j owns features [8j,8j+8) (== head j's slice).
// Computes logits, LeakyReLU, stores them, atomic segment-max (int key).
__global__ void __launch_bounds__(256)
edge_logits_kernel(const int* __restrict__ src_idx, const int* __restrict__ dst_idx,
                   const float* __restrict__ P_edge, const float* __restrict__ deter,
                   const float* __restrict__ dst_proj, const float* __restrict__ src_proj,
                   const float* __restrict__ W3, const float* __restrict__ W4,
                   float* __restrict__ logits_buf, int* __restrict__ seg_max_i, int E_)
{
    __shared__ float sW4[NHEAD * DIM];
    __shared__ float sW3[DIM];
    for (int i = threadIdx.x; i < NHEAD * DIM; i += blockDim.x) sW4[i] = W4[i];
    for (int i = threadIdx.x; i < DIM; i += blockDim.x)          sW3[i] = W3[i];
    __syncthreads();

    int gid = blockIdx.x * blockDim.x + threadIdx.x;
    int e = gid >> 3;
    int j = gid & 7;
    if (e >= E_) return;

    int   s  = src_idx[e];
    int   d  = dst_idx[e];
    float p  = P_edge[e];
    float dt = deter[e];

    int f0 = j * 8;
    const float4* dp = (const float4*)(dst_proj + (size_t)d * DIM + f0);
    const float4* sp = (const float4*)(src_proj + (size_t)s * DIM + f0);
    float4 a0 = dp[0], a1 = dp[1], b0 = sp[0], b1 = sp[1];

    float eb[8];
    eb[0] = a0.x + b0.x; eb[1] = a0.y + b0.y; eb[2] = a0.z + b0.z; eb[3] = a0.w + b0.w;
    eb[4] = a1.x + b1.x; eb[5] = a1.y + b1.y; eb[6] = a1.z + b1.z; eb[7] = a1.w + b1.w;
#pragma unroll
    for (int t = 0; t < 8; ++t) eb[t] += p * sW3[f0 + t];

    float part[NHEAD];
#pragma unroll
    for (int h = 0; h < NHEAD; ++h) {
        float acc = 0.0f;
#pragma unroll
        for (int t = 0; t < 8; ++t) acc += eb[t] * sW4[h * DIM + f0 + t];
        part[h] = acc;
    }
    // reduce across the 8 lanes of this edge (aligned groups, masks stay in-group)
#pragma unroll
    for (int mask = 1; mask < 8; mask <<= 1) {
#pragma unroll
        for (int h = 0; h < NHEAD; ++h)
            part[h] += __shfl_xor(part[h], mask, 32);
    }

    float lg = part[j] + dt;
    lg = (lg >= 0.0f) ? lg : 0.2f * lg;       // LeakyReLU(0.2)
    logits_buf[(size_t)e * NHEAD + j] = lg;

    int bi  = __float_as_int(lg);
    int key = (bi < 0) ? (bi ^ 0x7fffffff) : bi;   // order-preserving map
    atomicMax(&seg_max_i[d * NHEAD + j], key);
}

// exp(logit - segmax) in-place + atomic segment sum
__global__ void __launch_bounds__(256)
edge_exp_kernel(const int* __restrict__ dst_idx, const int* __restrict__ seg_max_i,
                float* __restrict__ buf, float* __restrict__ seg_sum, int E_)
{
    int idx = blockIdx.x * blockDim.x + threadIdx.x;
    if (idx >= E_ * NHEAD) return;
    int e = idx >> 3, h = idx & 7;
    int d = dst_idx[e];
    int key = seg_max_i[d * NHEAD + h];
    int bi  = (key < 0) ? (key ^ 0x7fffffff) : key;
    float mx = __int_as_float(bi);
    float ex = __expf(buf[idx] - mx);
    buf[idx] = ex;
    atomicAdd(&seg_sum[d * NHEAD + h], ex);
}

// alpha-weighted scatter-add of V into agg (8 lanes/edge, lane j = head j slice)
__global__ void __launch_bounds__(256)
edge_agg_kernel(const int* __restrict__ src_idx, const int* __restrict__ dst_idx,
                const float* __restrict__ buf, const float* __restrict__ seg_sum,
                const float* __restrict__ V, float* __restrict__ agg, int E_)
{
    int idx = blockIdx.x * blockDim.x + threadIdx.x;
    if (idx >= E_ * NHEAD) return;
    int e = idx >> 3, j = idx & 7;
    int s = src_idx[e], d = dst_idx[e];
    float alpha = buf[idx] / (seg_sum[d * NHEAD + j] + 1e-12f);

    const float4* vp = (const float4*)(V + (size_t)s * DIM + j * 8);
    float4 v0 = vp[0], v1 = vp[1];
    float* ap = agg + (size_t)d * DIM + j * 8;
    atomicAdd(ap + 0, alpha * v0.x); atomicAdd(ap + 1, alpha * v0.y);
    atomicAdd(ap + 2, alpha * v0.z); atomicAdd(ap + 3, alpha * v0.w);
    atomicAdd(ap + 4, alpha * v1.x); atomicAdd(ap + 5, alpha * v1.y);
    atomicAdd(ap + 6, alpha * v1.z); atomicAdd(ap + 7, alpha * v1.w);
}

// out = LN( agg@Wout.T + b_out + resid ) : WMMA + fused epilogue,
// LayerNorm stats via cross-lane shuffles (rows live across 16-lane halves).
__global__ void __launch_bounds__(128)
out_ln_kernel(const float* __restrict__ agg, const float* __restrict__ Wout,
              const float* __restrict__ b_out, const float* __restrict__ resid,
              const float* __restrict__ ln_g, const float* __restrict__ ln_b,
              float* __restrict__ out, int n)
{
    int tile = blockIdx.x * (blockDim.x >> 5) + (threadIdx.x >> 5);
    int r0 = tile * 16;
    if (r0 >= n) return;

    const int lane = threadIdx.x & 31;
    const int half = lane >> 4;
    const int m    = lane & 15;

    int rm = r0 + m; if (rm >= n) rm = n - 1;
    const float* arow = agg + (size_t)rm * DIM + half * 2;
    v2f a[16];
#pragma unroll
    for (int kb = 0; kb < 16; ++kb)
        a[kb] = *(const v2f*)(arow + kb * 4);

    float val[4][8];
#pragma unroll
    for (int nb = 0; nb < 4; ++nb) {
        const float* wrow = Wout + (size_t)(nb * 16 + m) * DIM + half * 2;
        v8f c = {};
#pragma unroll
        for (int kb = 0; kb < 16; ++kb) {
            v2f b = *(const v2f*)(wrow + kb * 4);
            c = __builtin_amdgcn_wmma_f32_16x16x4_f32(
                    false, a[kb], false, b, (short)0, c, false, false);
        }
        float bv = b_out[nb * 16 + m];
#pragma unroll
        for (int r = 0; r < 8; ++r) {
            int row = r0 + half * 8 + r;
            if (row >= n) row = n - 1;
            val[nb][r] = c[r] + bv + resid[(size_t)row * DIM + nb * 16 + m];
        }
    }

#pragma unroll
    for (int r = 0; r < 8; ++r) {
        float s = val[0][r] + val[1][r] + val[2][r] + val[3][r];
#pragma unroll
        for (int mask = 1; mask < 16; mask <<= 1) s += __shfl_xor(s, mask, 32);
        float mu = s * (1.0f / 64.0f);
        float q = 0.0f;
#pragma unroll
        for (int nb = 0; nb < 4; ++nb) { float dl = val[nb][r] - mu; q += dl * dl; }
#pragma unroll
        for (int mask = 1; mask < 16; mask <<= 1) q += __shfl_xor(q, mask, 32);
        float inv = rsqrtf(q * (1.0f / 64.0f) + 1e-5f);
        int row = r0 + half * 8 + r;
        if (row < n) {
#pragma unroll
            for (int nb = 0; nb < 4; ++nb) {
                int col = nb * 16 + m;
                out[(size_t)row * DIM + col] =
                    ln_g[col] * (val[nb][r] - mu) * inv + ln_b[col];
            }
        }
    }
}

extern "C" void kernel_launch(void* const* d_in, const int* in_sizes, int n_in,
                              void* d_out, int out_size, void* d_ws, size_t ws_size,
                              hipStream_t stream) {
    const float* dst_feats = (const float*)d_in[0];
    const float* src_feats = (const float*)d_in[1];
    const int*   edge_idx  = (const int*)  d_in[2];
    const float* P_edge    = (const float*)d_in[3];
    const float* deter     = (const float*)d_in[4];
    const float* W1        = (const float*)d_in[5];
    const float* W2        = (const float*)d_in[6];
    const float* W3        = (const float*)d_in[7];
    const float* W4        = (const float*)d_in[8];
    const float* Wv        = (const float*)d_in[9];
    const float* Wout      = (const float*)d_in[10];
    const float* b_out     = (const float*)d_in[11];
    const float* Wres      = (const float*)d_in[12];
    const float* b_res     = (const float*)d_in[13];
    const float* ln_g      = (const float*)d_in[14];
    const float* ln_b      = (const float*)d_in[15];
    float* out = (float*)d_out;

    const int E_    = in_sizes[3];
    const int n_dst = in_sizes[0] / DIM;
    const int n_src = in_sizes[1] / DIM;
    const int* src_idx = edge_idx;
    const int* dst_idx = edge_idx + E_;

    char* ws = (char*)d_ws;
    float* src_proj  = (float*)ws; ws += (size_t)n_src * DIM * 4;
    float* Vbuf      = (float*)ws; ws += (size_t)n_src * DIM * 4;
    float* dst_proj  = (float*)ws; ws += (size_t)n_dst * DIM * 4;
    float* residbuf  = (float*)ws; ws += (size_t)n_dst * DIM * 4;
    float* agg       = (float*)ws; ws += (size_t)n_dst * DIM * 4;
    float* seg_sum   = (float*)ws; ws += (size_t)n_dst * NHEAD * 4;
    int*   seg_max_i = (int*)  ws; ws += (size_t)n_dst * NHEAD * 4;
    float* logits    = (float*)ws; ws += (size_t)E_ * NHEAD * 4;

    {   // node projections (WMMA): 4 waves/block, 16 rows/wave
        int tiles  = (n_src + 15) / 16;
        int blocks = (tiles + 3) / 4;
        node_proj_src_kernel<<<blocks, 128, 0, stream>>>(
            src_feats, W2, Wv, src_proj, Vbuf, n_src);
    }
    {
        int tiles  = (n_dst + 15) / 16;
        int blocks = (tiles + 3) / 4;
        node_proj_dst_kernel<<<blocks, 128, 0, stream>>>(
            dst_feats, W1, Wres, b_res, dst_proj, residbuf, n_dst);
    }
    {
        int total  = n_dst * DIM;
        int blocks = (total + 255) / 256;
        init_kernel<<<blocks, 256, 0, stream>>>(agg, seg_max_i, seg_sum, n_dst);
    }
    {
        long long total = (long long)E_ * NHEAD;
        int blocks = (int)((total + 255) / 256);
        edge_logits_kernel<<<blocks, 256, 0, stream>>>(
            src_idx, dst_idx, P_edge, deter, dst_proj, src_proj,
            W3, W4, logits, seg_max_i, E_);
        edge_exp_kernel<<<blocks, 256, 0, stream>>>(
            dst_idx, seg_max_i, logits, seg_sum, E_);
        edge_agg_kernel<<<blocks, 256, 0, stream>>>(
            src_idx, dst_idx, logits, seg_sum, Vbuf, agg, E_);
    }
    {
        int tiles  = (n_dst + 15) / 16;
        int blocks = (tiles + 3) / 4;
        out_ln_kernel<<<blocks, 128, 0, stream>>>(
            agg, Wout, b_out, residbuf, ln_g, ln_b, out, n_dst);
    }
}